// PatchCore_2585570312716
// MI455X (gfx1250) — compile-verified
//
#include <hip/hip_runtime.h>

#define N_PATCHES 8192
#define MEM_SIZE  32768
#define FEAT      512

#define NT   64    // block rows (patches)
#define MTT  128   // block cols (memory bank)
#define KC   64    // K chunk staged in LDS
#define LDSS 68    // padded LDS row stride (floats): 272B = 17*16B -> float4 aligned, conflict-free
#define NCHUNK (FEAT / KC)

typedef __attribute__((ext_vector_type(2))) float v2f;
typedef __attribute__((ext_vector_type(8))) float v8f;

// Pointee type exactly as the builtin expects: vector_size(16) int
typedef int i32x4 __attribute__((vector_size(16)));
typedef __attribute__((address_space(1))) i32x4 gi4;   // global int4
typedef __attribute__((address_space(3))) i32x4 li4;   // LDS int4

// ---- CDNA5 async global->LDS copy (ASYNCcnt-tracked, no VGPR staging) ------
__device__ __forceinline__ void async_copy_b128(const float* g, float* l) {
#if defined(__has_builtin) && __has_builtin(__builtin_amdgcn_global_load_async_to_lds_b128)
    __builtin_amdgcn_global_load_async_to_lds_b128((gi4*)g, (li4*)l, 0, 0);
#else
    const unsigned loff = (unsigned)(unsigned long long)(li4*)l;  // 32-bit LDS offset
    asm volatile("global_load_async_to_lds_b128 %0, %1, off"
                 :: "v"(loff), "v"((unsigned long long)g) : "memory");
#endif
}

__device__ __forceinline__ void wait_async0() {
#if defined(__has_builtin) && __has_builtin(__builtin_amdgcn_s_wait_asynccnt)
    __builtin_amdgcn_s_wait_asynccnt(0);
#else
    asm volatile("s_wait_asynccnt 0" ::: "memory");
#endif
}

// ---------------------------------------------------------------------------
// Kernel 1: squared norms of every row of patches and memory bank.
// One wave per row; coalesced float4 loads; wave shfl reduction.
// Also initializes the per-patch min array to +inf.
// ---------------------------------------------------------------------------
__global__ __launch_bounds__(256)
void sq_init_kernel(const float* __restrict__ patches,
                    const float* __restrict__ mem,
                    float* __restrict__ p_sq,
                    float* __restrict__ m_sq,
                    unsigned* __restrict__ minarr)
{
    const int wave = (int)((blockIdx.x * blockDim.x + threadIdx.x) >> 5);
    const int lane = (int)(threadIdx.x & 31);
    if (wave >= N_PATCHES + MEM_SIZE) return;

    const float* row = (wave < N_PATCHES)
        ? (patches + (size_t)wave * FEAT)
        : (mem + (size_t)(wave - N_PATCHES) * FEAT);

    float s = 0.0f;
#pragma unroll
    for (int j = 0; j < 4; ++j) {
        const float4 v = *(const float4*)(row + lane * 4 + j * 128);
        s = fmaf(v.x, v.x, fmaf(v.y, v.y, fmaf(v.z, v.z, fmaf(v.w, v.w, s))));
    }
#pragma unroll
    for (int off = 16; off; off >>= 1) s += __shfl_xor(s, off, 32);

    if (lane == 0) {
        if (wave < N_PATCHES) {
            p_sq[wave] = s;
            minarr[wave] = 0x7F800000u;   // +inf
        } else {
            m_sq[wave - N_PATCHES] = s;
        }
    }
}

// ---------------------------------------------------------------------------
// Kernel 2: double-buffered, async-staged WMMA GEMM fused with row-min.
// Block tile 64(N) x 128(M); 8 waves in 2x4; each wave owns 32x32
// (four 16x16 f32 WMMA accumulators, full A/B fragment reuse).
// ---------------------------------------------------------------------------
__global__ __launch_bounds__(256)
void gemm_min_kernel(const float* __restrict__ A,      // patches  [8192 x 512]
                     const float* __restrict__ B,      // memory   [32768 x 512]
                     const float* __restrict__ p_sq,
                     const float* __restrict__ m_sq,
                     unsigned* __restrict__ minarr)
{
    __shared__ float sA[2][NT  * LDSS];   //  ~34.8 KB
    __shared__ float sB[2][MTT * LDSS];   //  ~69.6 KB

    const int nbase = (int)blockIdx.x * NT;    // gridDim.x = 8192/64  = 128
    const int mbase = (int)blockIdx.y * MTT;   // gridDim.y = 32768/128 = 256
    const int t     = (int)threadIdx.x;
    const int lane  = t & 31;
    const int wave  = t >> 5;
    const int wr    = wave & 1;                // 0..1 -> 32-row strip
    const int wc    = wave >> 1;               // 0..3 -> 32-col strip

    const int m16   = lane & 15;
    const int khalf = lane >> 4;               // K split across lane halves

    v8f c00 = {}, c01 = {}, c10 = {}, c11 = {};

    // ---- async stage of one K-chunk into buffer `buf` ----
    auto stage = [&](int buf, int k0) {
#pragma unroll
        for (int j = 0; j < 4; ++j) {          // A: 64x64 f32 = 1024 float4
            const int f    = t + 256 * j;
            const int row  = f >> 4;
            const int colv = f & 15;
            async_copy_b128(A + (size_t)(nbase + row) * FEAT + k0 + colv * 4,
                            &sA[buf][row * LDSS + colv * 4]);
        }
#pragma unroll
        for (int j = 0; j < 8; ++j) {          // B: 128x64 f32 = 2048 float4
            const int f    = t + 256 * j;
            const int row  = f >> 4;
            const int colv = f & 15;
            async_copy_b128(B + (size_t)(mbase + row) * FEAT + k0 + colv * 4,
                            &sB[buf][row * LDSS + colv * 4]);
        }
    };

    // ---- WMMA sweep over one staged K-chunk ----
    auto compute = [&](int buf) {
        const float* __restrict__ bufA = sA[buf];
        const float* __restrict__ bufB = sB[buf];
#pragma unroll
        for (int kk = 0; kk < KC; kk += 4) {
            const v2f a0 = *(const v2f*)(bufA + (wr * 32 +      m16) * LDSS + kk + 2 * khalf);
            const v2f a1 = *(const v2f*)(bufA + (wr * 32 + 16 + m16) * LDSS + kk + 2 * khalf);
            const v2f b0 = *(const v2f*)(bufB + (wc * 32 +      m16) * LDSS + kk + 2 * khalf);
            const v2f b1 = *(const v2f*)(bufB + (wc * 32 + 16 + m16) * LDSS + kk + 2 * khalf);
            c00 = __builtin_amdgcn_wmma_f32_16x16x4_f32(false, a0, false, b0, (short)0, c00, false, false);
            c01 = __builtin_amdgcn_wmma_f32_16x16x4_f32(false, a0, false, b1, (short)0, c01, false, false);
            c10 = __builtin_amdgcn_wmma_f32_16x16x4_f32(false, a1, false, b0, (short)0, c10, false, false);
            c11 = __builtin_amdgcn_wmma_f32_16x16x4_f32(false, a1, false, b1, (short)0, c11, false, false);
        }
    };

    // ---- double-buffered main loop: prefetch chunk c+1 while computing c ----
    stage(0, 0);
    int cur = 0;
#pragma unroll 1
    for (int c = 0; c < NCHUNK; ++c) {
        wait_async0();        // my async transfers into buf[cur] are done
        __syncthreads();      // everyone's are -> buf[cur] fully populated
        if (c + 1 < NCHUNK) stage(cur ^ 1, (c + 1) * KC);
        compute(cur);
        cur ^= 1;
    }

    // ---- epilogue: d2' = m_sq[c] - 2*cross; min over this wave's 32 cols;
    //      + p_sq[row]; clamp >= 0; uint-ordered atomic min (deterministic) ----
    const float msq0 = m_sq[mbase + wc * 32 +      m16];
    const float msq1 = m_sq[mbase + wc * 32 + 16 + m16];

#pragma unroll
    for (int i = 0; i < 8; ++i) {
        // C layout: VGPR i -> row i (lanes 0-15) / row i+8 (lanes 16-31); col = lane&15
        float v = fminf(fmaf(-2.0f, c00[i], msq0), fmaf(-2.0f, c01[i], msq1));
        v = fminf(v, __shfl_xor(v, 1, 32));
        v = fminf(v, __shfl_xor(v, 2, 32));
        v = fminf(v, __shfl_xor(v, 4, 32));
        v = fminf(v, __shfl_xor(v, 8, 32));
        float w = fminf(fmaf(-2.0f, c10[i], msq0), fmaf(-2.0f, c11[i], msq1));
        w = fminf(w, __shfl_xor(w, 1, 32));
        w = fminf(w, __shfl_xor(w, 2, 32));
        w = fminf(w, __shfl_xor(w, 4, 32));
        w = fminf(w, __shfl_xor(w, 8, 32));
        if (m16 == 0) {
            const int r0 = nbase + wr * 32 + i + 8 * khalf;
            const float d0 = fmaxf(v + p_sq[r0], 0.0f);
            atomicMin(minarr + r0, __float_as_uint(d0));
            const int r1 = r0 + 16;
            const float d1 = fmaxf(w + p_sq[r1], 0.0f);
            atomicMin(minarr + r1, __float_as_uint(d1));
        }
    }
}

// ---------------------------------------------------------------------------
// Kernel 3: max over the 8192 per-patch minima (uint order == float order for
// non-negative values), then sqrt -> scalar output.
// ---------------------------------------------------------------------------
__global__ __launch_bounds__(256)
void max_reduce_kernel(const unsigned* __restrict__ minarr, float* __restrict__ out)
{
    __shared__ unsigned smax[8];
    unsigned m = 0u;
    for (int i = (int)threadIdx.x; i < N_PATCHES; i += 256) {
        const unsigned v = minarr[i];
        m = (v > m) ? v : m;
    }
#pragma unroll
    for (int off = 16; off; off >>= 1) {
        const unsigned o = __shfl_xor(m, off, 32);
        m = (o > m) ? o : m;
    }
    if ((threadIdx.x & 31) == 0) smax[threadIdx.x >> 5] = m;
    __syncthreads();
    if (threadIdx.x == 0) {
        unsigned r = smax[0];
#pragma unroll
        for (int i = 1; i < 8; ++i) r = (smax[i] > r) ? smax[i] : r;
        out[0] = sqrtf(__uint_as_float(r));
    }
}

// ---------------------------------------------------------------------------
extern "C" void kernel_launch(void* const* d_in, const int* in_sizes, int n_in,
                              void* d_out, int out_size, void* d_ws, size_t ws_size,
                              hipStream_t stream) {
    (void)in_sizes; (void)n_in; (void)out_size; (void)ws_size;
    const float* patches = (const float*)d_in[0];   // [8192, 512] f32
    const float* membank = (const float*)d_in[1];   // [32768, 512] f32
    float* out = (float*)d_out;                     // scalar f32

    // Workspace layout (f32 elements): p_sq[8192] | m_sq[32768] | minarr[8192]
    float*    p_sq   = (float*)d_ws;
    float*    m_sq   = p_sq + N_PATCHES;
    unsigned* minarr = (unsigned*)(m_sq + MEM_SIZE);

    {
        const int waves  = N_PATCHES + MEM_SIZE;
        const int blocks = (waves * 32 + 255) / 256;
        sq_init_kernel<<<blocks, 256, 0, stream>>>(patches, membank, p_sq, m_sq, minarr);
    }
    {
        dim3 grid(N_PATCHES / NT, MEM_SIZE / MTT);   // 128 x 256
        gemm_min_kernel<<<grid, 256, 0, stream>>>(patches, membank, p_sq, m_sq, minarr);
    }
    max_reduce_kernel<<<1, 256, 0, stream>>>(minarr, out);
}